// MixSelfAttention_88046829568163
// MI455X (gfx1250) — compile-verified
//
#include <hip/hip_runtime.h>
#include <math.h>

typedef float v2f __attribute__((ext_vector_type(2)));
typedef float v8f __attribute__((ext_vector_type(8)));

#define B_ 4
#define L_ 1024
#define H_ 8
#define E_ 64
#define NTOP 35
#define SCALE_ 0.04419417382415922f  // 1/sqrt(512)

// ---------------------------------------------------------------------------
// Kernel 1a: partial correlation energy for one 8-wide e-chunk.
//   pamp[bh][ec][tau] = sum_{e in chunk} ( sum_t q[t,e]*k[(t-tau) mod L, e] )^2
// Each thread owns 4 adjacent tau; a 3-deep register window re-uses each k
// value across the 4 accumulators -> ~0.31 LDS ops per FMA (VALU-bound).
// Grid: (32 bh, 8 e-chunks), block 256.
// ---------------------------------------------------------------------------
__global__ __launch_bounds__(256) void corr_partial_kernel(
    const float* __restrict__ q, const float* __restrict__ k,
    float* __restrict__ pamp) {
  const int bh = blockIdx.x;
  const int b = bh >> 3, h = bh & 7;
  const int ec = blockIdx.y;
  const int tid = threadIdx.x;
  const int tau0 = tid << 2;  // 4 adjacent taus per thread

  __shared__ float qs[8][1032];   // padded rows: bank-shift 8 per e
  __shared__ float ks[8][2056];   // k duplicated to 2L: wrap-free window

  const int ebase = ec << 3;
  for (int i = tid; i < 8192; i += 256) {
    const int t = i >> 3, j = i & 7;
    const size_t g = (((size_t)b * L_ + t) * H_ + h) * E_ + ebase + j;
    const float qv = q[g];
    const float kv = k[g];
    qs[j][t] = qv;
    ks[j][t] = kv;
    ks[j][t + 1024] = kv;
  }
  __syncthreads();

  float s0 = 0.0f, s1 = 0.0f, s2 = 0.0f, s3 = 0.0f;
  const int w0 = 1024 - tau0;  // base k-index; w0-3 >= 1 for tau0 <= 1020
#pragma unroll
  for (int j = 0; j < 8; ++j) {
    const float* __restrict__ qr = &qs[j][0];   // wave-uniform (broadcast)
    const float* __restrict__ kr = &ks[j][w0];  // per-lane window
    float p1 = kr[-1], p2 = kr[-2], p3 = kr[-3];
    float a0 = 0.0f, a1 = 0.0f, a2 = 0.0f, a3 = 0.0f;
#pragma unroll 4
    for (int t = 0; t < 1024; ++t) {
      const float cur = kr[t];
      const float qv = qr[t];
      a0 = fmaf(qv, cur, a0);  // tau0+0 uses k[t - tau0]
      a1 = fmaf(qv, p1, a1);   // tau0+1 uses k[t - tau0 - 1]
      a2 = fmaf(qv, p2, a2);
      a3 = fmaf(qv, p3, a3);
      p3 = p2; p2 = p1; p1 = cur;
    }
    s0 = fmaf(a0, a0, s0);
    s1 = fmaf(a1, a1, s1);
    s2 = fmaf(a2, a2, s2);
    s3 = fmaf(a3, a3, s3);
  }
  float4 r; r.x = s0; r.y = s1; r.z = s2; r.w = s3;
  *(float4*)&pamp[((size_t)bh * 8 + ec) * L_ + tau0] = r;
}

// ---------------------------------------------------------------------------
// Kernel 1b: deterministic reduce of the 8 e-chunk partials + sqrt.
// ---------------------------------------------------------------------------
__global__ __launch_bounds__(256) void corr_reduce_kernel(
    const float* __restrict__ pamp, float* __restrict__ amp) {
  const int bh = blockIdx.x;
  const int tau = (blockIdx.y << 8) + threadIdx.x;
  const float* p = &pamp[(size_t)bh * 8 * L_ + tau];
  float s = 0.0f;
#pragma unroll
  for (int ec = 0; ec < 8; ++ec) s += p[ec * L_];  // fixed order: deterministic
  amp[bh * L_ + tau] = sqrtf(s);
}

// ---------------------------------------------------------------------------
// Kernel 2: amp_tf[bh][tau] = || tfq[b,tau,h,:] ||   (corr=False path)
// ---------------------------------------------------------------------------
__global__ __launch_bounds__(256) void tf_amp_kernel(
    const float* __restrict__ tfq, float* __restrict__ amp) {
  const int bh = blockIdx.x;
  const int b = bh >> 3, h = bh & 7;
  const int tau = (blockIdx.y << 8) + threadIdx.x;
  const float4* p = (const float4*)(tfq + (((size_t)b * L_ + tau) * H_ + h) * E_);
  float s = 0.0f;
#pragma unroll
  for (int i = 0; i < 16; ++i) {
    const float4 x = p[i];
    s = fmaf(x.x, x.x, s); s = fmaf(x.y, x.y, s);
    s = fmaf(x.z, x.z, s); s = fmaf(x.w, x.w, s);
  }
  amp[bh * L_ + tau] = sqrtf(s);
}

// ---------------------------------------------------------------------------
// Kernel 3: deterministic top-35 indices (35 rounds of argmax, index
// tie-break). Only the index SET matters downstream.
// ---------------------------------------------------------------------------
__global__ __launch_bounds__(256) void topk_kernel(
    const float* __restrict__ ampT, const float* __restrict__ ampF,
    int* __restrict__ idxT, int* __restrict__ idxF) {
  const int bh = blockIdx.x;
  const int path = blockIdx.y;
  const float* amp = (path == 0 ? ampT : ampF) + bh * L_;
  int* idx = (path == 0 ? idxT : idxF) + bh * 64;
  const int tid = threadIdx.x;

  __shared__ float vals[1024];
  __shared__ float rv[256];
  __shared__ int   ri[256];

  for (int i = tid; i < 1024; i += 256) vals[i] = amp[i];
  __syncthreads();

  for (int it = 0; it < NTOP; ++it) {
    float bv = -1e30f; int bi = 0;
#pragma unroll
    for (int j = 0; j < 4; ++j) {
      const int i = tid + (j << 8);
      const float vx = vals[i];
      if (vx > bv) { bv = vx; bi = i; }
    }
    rv[tid] = bv; ri[tid] = bi;
    __syncthreads();
    for (int s = 128; s > 0; s >>= 1) {
      if (tid < s) {
        const float ov = rv[tid + s]; const int oi = ri[tid + s];
        if (ov > rv[tid] || (ov == rv[tid] && oi < ri[tid])) {
          rv[tid] = ov; ri[tid] = oi;
        }
      }
      __syncthreads();
    }
    if (tid == 0) {
      idx[it] = ri[0];
      vals[ri[0]] = -1e30f;
    }
    __syncthreads();
  }
}

// ---------------------------------------------------------------------------
// Kernel 4: fused scores -> softmax -> mix-weight -> value contraction,
// all matmuls on V_WMMA_F32_16X16X4_F32 (fp32 exact).
// Grid: (32 bh, 8 key-tiles of 128). Block: 256 (8 waves).
// ---------------------------------------------------------------------------
#define QSTR 68
#define VSTR 68
#define KSTR 68
#define SSTR 128
#define OSTR 132

__global__ __launch_bounds__(256) void attn_out_kernel(
    const float* __restrict__ q, const float* __restrict__ k,
    const float* __restrict__ tfq, const float* __restrict__ v,
    const float* __restrict__ mix,
    const int* __restrict__ idxT, const int* __restrict__ idxF,
    float* __restrict__ out) {
  const int bh = blockIdx.x;
  const int b = bh >> 3, h = bh & 7;
  const int lk0 = blockIdx.y << 7;
  const int tid = threadIdx.x;
  const int lane = tid & 31;
  const int wid = tid >> 5;
  const int l15 = lane & 15;
  const int l16 = (lane >> 4) & 1;

  __shared__ float Qt[48 * QSTR];   // gathered queries (path t), rows 35..47 = 0
  __shared__ float Qf[48 * QSTR];   // gathered tf_queries (path tf)
  __shared__ float Vs[96 * VSTR];   // stacked gathered values: [0,48)=t, [48,96)=tf
  __shared__ float Ks[128 * KSTR];  // key tile; reused for tfq tile, then O staging
  __shared__ float Ss[96 * SSTR];   // scores -> probabilities (in place)
  __shared__ int   sidx[96];

  if (tid < 96) {
    const int set = tid >= 48;
    const int r = tid - 48 * set;
    sidx[tid] = (r < NTOP) ? (set ? idxF : idxT)[bh * 64 + r] : 0;
  }
  __syncthreads();

  // gather selected queries and values (zero-pad rows >= 35)
  for (int i = tid; i < 48 * 64; i += 256) {
    const int r = i >> 6, e = i & 63;
    float qt = 0.0f, qf = 0.0f;
    if (r < NTOP) {
      qt = q[(((size_t)b * L_ + sidx[r]) * H_ + h) * E_ + e];
      qf = tfq[(((size_t)b * L_ + sidx[48 + r]) * H_ + h) * E_ + e];
    }
    Qt[r * QSTR + e] = qt;
    Qf[r * QSTR + e] = qf;
  }
  for (int i = tid; i < 96 * 64; i += 256) {
    const int r = i >> 6, e = i & 63;
    const int set = r >= 48;
    const int rr = r - 48 * set;
    float vv = 0.0f;
    if (rr < NTOP)
      vv = v[(((size_t)b * L_ + sidx[set * 48 + rr]) * H_ + h) * E_ + e];
    Vs[r * VSTR + e] = vv;
  }
  for (int i = tid; i < 128 * 64; i += 256) {
    const int r = i >> 6, e = i & 63;
    Ks[r * KSTR + e] = k[(((size_t)b * L_ + lk0 + r) * H_ + h) * E_ + e];
  }
  __syncthreads();

  // ---- S = Q_sel . K^T  (M=48 pad, N=128, K=64) : 3x8 tiles over 8 waves,
  //      two passes through one key-tile buffer (keys, then tf_queries)
#pragma unroll
  for (int pass = 0; pass < 2; ++pass) {
    const float* Qm = pass ? Qf : Qt;
    for (int it = 0; it < 3; ++it) {
      const int m0 = it << 4;        // 0,16,32
      const int n0 = wid << 4;       // key sub-tile
      v8f c = {0.f, 0.f, 0.f, 0.f, 0.f, 0.f, 0.f, 0.f};
#pragma unroll
      for (int kc = 0; kc < 16; ++kc) {
        const int kb = (kc << 2) + (l16 << 1);
        v2f a, bb;
        a[0] = Qm[(m0 + l15) * QSTR + kb];      // A[m,k],A[m,k+1]
        a[1] = Qm[(m0 + l15) * QSTR + kb + 1];
        bb[0] = Ks[(n0 + l15) * KSTR + kb];     // B[k,n] = K[lk=n][e=k]
        bb[1] = Ks[(n0 + l15) * KSTR + kb + 1];
        c = __builtin_amdgcn_wmma_f32_16x16x4_f32(false, a, false, bb,
                                                  (short)0, c, false, false);
      }
      const int sm = pass * 48 + m0 + (l16 << 3);
#pragma unroll
      for (int r = 0; r < 8; ++r) Ss[(sm + r) * SSTR + n0 + l15] = c[r];
    }
    __syncthreads();
    if (pass == 0) {
      // swap the tile buffer to tf_queries for the second pass
      for (int i = tid; i < 128 * 64; i += 256) {
        const int r = i >> 6, e = i & 63;
        Ks[r * KSTR + e] = tfq[(((size_t)b * L_ + lk0 + r) * H_ + h) * E_ + e];
      }
      __syncthreads();
    }
  }

  // ---- per-key softmax over the 35 rows; fold mix weight into P; zero pads
  {
    const int col = tid & 127;
    const int path = tid >> 7;
    const int rb = path * 48;
    const int lk = lk0 + col;
    const float mwa = mix[(h * L_ + lk) * 2 + 0];
    const float mwb = mix[(h * L_ + lk) * 2 + 1];
    const float mm = fmaxf(mwa, mwb);
    const float ea = expf(mwa - mm), eb = expf(mwb - mm);
    const float w = (path == 0 ? ea : eb) / (ea + eb);

    float mx = -1e30f;
    for (int n = 0; n < NTOP; ++n)
      mx = fmaxf(mx, Ss[(rb + n) * SSTR + col] * SCALE_);
    float den = 0.0f;
    for (int n = 0; n < NTOP; ++n) {
      const float e_ = expf(Ss[(rb + n) * SSTR + col] * SCALE_ - mx);
      den += e_;
      Ss[(rb + n) * SSTR + col] = e_;
    }
    const float inv = w / den;
    for (int n = 0; n < NTOP; ++n) Ss[(rb + n) * SSTR + col] *= inv;
    for (int n = NTOP; n < 48; ++n) Ss[(rb + n) * SSTR + col] = 0.0f;
  }
  __syncthreads();

  // ---- O[lk,e] = sum_n P[n,lk] * V[n,e]  (M=128, N=64, K=96 stacked paths)
#pragma unroll
  for (int nt = 0; nt < 4; ++nt) {
    const int m0 = wid << 4;   // lk sub-tile
    const int e0 = nt << 4;    // e sub-tile
    v8f c = {0.f, 0.f, 0.f, 0.f, 0.f, 0.f, 0.f, 0.f};
#pragma unroll
    for (int kc = 0; kc < 24; ++kc) {
      const int kr = (kc << 2) + (l16 << 1);
      v2f a, bb;
      a[0] = Ss[kr * SSTR + m0 + l15];          // A[m=lk,k=n] = P[n][lk]
      a[1] = Ss[(kr + 1) * SSTR + m0 + l15];
      bb[0] = Vs[kr * VSTR + e0 + l15];         // B[k=n,n=e]
      bb[1] = Vs[(kr + 1) * VSTR + e0 + l15];
      c = __builtin_amdgcn_wmma_f32_16x16x4_f32(false, a, false, bb,
                                                (short)0, c, false, false);
    }
    // stage transposed into the (now free) Ks buffer: Os[e][lk_local]
#pragma unroll
    for (int r = 0; r < 8; ++r)
      Ks[(e0 + l15) * OSTR + m0 + (l16 << 3) + r] = c[r];
  }
  __syncthreads();

  // ---- coalesced writeout: out[b,h,e,lk]
  for (int i = tid; i < 64 * 128; i += 256) {
    const int e = i >> 7, cc = i & 127;
    out[(((size_t)b * H_ + h) * E_ + e) * L_ + lk0 + cc] = Ks[e * OSTR + cc];
  }
}

// ---------------------------------------------------------------------------
extern "C" void kernel_launch(void* const* d_in, const int* in_sizes, int n_in,
                              void* d_out, int out_size, void* d_ws, size_t ws_size,
                              hipStream_t stream) {
  (void)in_sizes; (void)n_in; (void)out_size; (void)ws_size;
  const float* tfq = (const float*)d_in[0];
  const float* q   = (const float*)d_in[1];
  const float* key = (const float*)d_in[2];
  const float* val = (const float*)d_in[3];
  const float* mix = (const float*)d_in[6];
  float* out = (float*)d_out;

  float* amp_t = (float*)d_ws;              // [32][1024]
  float* amp_f = amp_t + 32 * 1024;         // [32][1024]
  int*   idx_t = (int*)(amp_f + 32 * 1024); // [32][64]
  int*   idx_f = idx_t + 32 * 64;           // [32][64]
  float* pamp  = (float*)(idx_f + 32 * 64); // [32][8][1024] partial energies

  corr_partial_kernel<<<dim3(32, 8), 256, 0, stream>>>(q, key, pamp);
  corr_reduce_kernel<<<dim3(32, 4), 256, 0, stream>>>(pamp, amp_t);
  tf_amp_kernel<<<dim3(32, 4), 256, 0, stream>>>(tfq, amp_f);
  topk_kernel<<<dim3(32, 2), 256, 0, stream>>>(amp_t, amp_f, idx_t, idx_f);
  attn_out_kernel<<<dim3(32, 8), 256, 0, stream>>>(q, key, tfq, val, mix,
                                                   idx_t, idx_f, out);
}